// OICRe2e_83923660963979
// MI455X (gfx1250) — compile-verified
//
#include <hip/hip_runtime.h>
#include <math.h>

#define BS 16
#define P 4096
#define MAX_GT 32
#define C_IN 512
#define NCLS 4

typedef __attribute__((ext_vector_type(2))) float v2f;
typedef __attribute__((ext_vector_type(8))) float v8f;

// ---------------------------------------------------------------------------
// Kernel 1: streaming GEMM  out[65536][8] = v[65536][512] @ [cls_w;reg_w]^T + b
// One wave per 16-row tile, K accumulated in chunks of 4 via WMMA f32 16x16x4.
// B operand staged in LDS as a 16x512 tile: rows 0..7 = [cls_w;reg_w],
// rows 8..15 = zeros, so the inner loop is pure loads + WMMA (no masking ALU).
// ---------------------------------------------------------------------------
__global__ __launch_bounds__(256) void gemm16(const float* __restrict__ v,
                                              const float* __restrict__ cls_w,
                                              const float* __restrict__ cls_b,
                                              const float* __restrict__ reg_w,
                                              const float* __restrict__ reg_b,
                                              float* __restrict__ out) {
  __shared__ float wlds[16 * C_IN];  // rows 0..3 cls, 4..7 reg, 8..15 zero
  const int tid = threadIdx.x;
  for (int i = tid; i < 4 * C_IN; i += 256) {
    wlds[i] = cls_w[i];
    wlds[4 * C_IN + i] = reg_w[i];
    wlds[8 * C_IN + i] = 0.0f;
    wlds[12 * C_IN + i] = 0.0f;
  }
  __syncthreads();

  const int wave = tid >> 5;
  const int lane = tid & 31;
  const int tile = blockIdx.x * 8 + wave;      // 0..4095
  const int row0 = tile * 16;
  const int m    = lane & 15;                  // A-matrix row within tile
  const int koff = (lane >> 4) << 1;           // K sub-offset: 0 or 2
  const int n    = lane & 15;                  // B-matrix column

  const float* arow = v + (size_t)(row0 + m) * C_IN + koff;
  const float* brow = wlds + n * C_IN + koff;

  v8f acc = {0.f, 0.f, 0.f, 0.f, 0.f, 0.f, 0.f, 0.f};
#pragma unroll 4
  for (int k0 = 0; k0 < C_IN; k0 += 4) {
    v2f a  = *(const v2f*)(arow + k0);         // global_load_b64
    v2f bf = *(const v2f*)(brow + k0);         // ds_load_b64 (pre-masked)
    acc = __builtin_amdgcn_wmma_f32_16x16x4_f32(
        /*neg_a=*/false, a, /*neg_b=*/false, bf,
        /*c_mod=*/(short)0, acc, /*reuse_a=*/false, /*reuse_b=*/false);
  }

  // C/D layout: VGPR j -> (M=j, N=lane) for lanes 0-15, (M=j+8, N=lane-16) 16-31
  if (n < 8) {
    const float bias = (n < 4) ? cls_b[n] : reg_b[n - 4];
    const int rbase = row0 + ((lane >> 4) << 3);
#pragma unroll
    for (int j = 0; j < 8; ++j) {
      out[(size_t)(rbase + j) * 8 + n] = acc[j] + bias;
    }
  }
}

// ---------------------------------------------------------------------------
// Kernel 2: per-batch loss terms -> 4 partial sums per batch
// ---------------------------------------------------------------------------
__device__ __forceinline__ float sl1(float x) {
  float ax = fabsf(x);
  return (ax < 1.0f) ? 0.5f * ax * ax : ax - 0.5f;
}

__global__ __launch_bounds__(256) void loss_per_batch(
    const float* __restrict__ cr,        // [BS*P][8]: 4 cls logits, 4 reg
    const float* __restrict__ gt_boxes,  // [BS][MAX_GT][4]
    const int*   __restrict__ gt_counts, // [BS]
    const float* __restrict__ rois,      // [BS][P][4]
    const float* __restrict__ labels,    // [BS][NCLS] one-hot
    const float* __restrict__ pre_score, // [BS][P][NCLS]
    float* __restrict__ part) {          // [BS][4]
  const int b = blockIdx.x;
  const int tid = threadIdx.x;
  __shared__ float sgt[MAX_GT * 4];
  __shared__ int s_lab;
  __shared__ float sred[256 * 4];

  if (tid < MAX_GT * 4) sgt[tid] = gt_boxes[b * MAX_GT * 4 + tid];
  if (tid == 0) {
    int lab = 0;
    float bestl = labels[b * NCLS];
    for (int j = 1; j < NCLS; ++j) {
      float lj = labels[b * NCLS + j];
      if (lj > bestl) { bestl = lj; lab = j; }
    }
    s_lab = lab;
  }
  __syncthreads();
  const int lab = s_lab;
  const int cnt = gt_counts[b];

  float acc_bg = 0.f, acc_fg = 0.f, acc_mask = 0.f, acc_l1 = 0.f;

  for (int p = tid; p < P; p += 256) {
    const size_t gp = (size_t)b * P + p;
    const float* cp = cr + gp * 8;
    float c0 = cp[0], c1 = cp[1], c2 = cp[2], c3 = cp[3];
    float r0 = cp[4], r1 = cp[5], r2 = cp[6], r3 = cp[7];

    // background CE: logsumexp(raw) - raw[label]
    float mx  = fmaxf(fmaxf(c0, c1), fmaxf(c2, c3));
    float lse = mx + logf(expf(c0 - mx) + expf(c1 - mx) +
                          expf(c2 - mx) + expf(c3 - mx));
    float clab = (lab == 0) ? c0 : (lab == 1) ? c1 : (lab == 2) ? c2 : c3;
    float ce_bg = lse - clab;

    // foreground CE on clamped logits, weighted by pre_score[..,label]
    const float lo = 1e-7f, hi = 1.0f - 1e-7f;
    float q0 = fminf(fmaxf(c0, lo), hi), q1 = fminf(fmaxf(c1, lo), hi);
    float q2 = fminf(fmaxf(c2, lo), hi), q3 = fminf(fmaxf(c3, lo), hi);
    float mq  = fmaxf(fmaxf(q0, q1), fmaxf(q2, q3));
    float lseq = mq + logf(expf(q0 - mq) + expf(q1 - mq) +
                           expf(q2 - mq) + expf(q3 - mq));
    float qlab = (lab == 0) ? q0 : (lab == 1) ? q1 : (lab == 2) ? q2 : q3;
    float ps = pre_score[gp * NCLS + lab];
    float ce_fg = (lseq - qlab) * ps;

    // roi
    const float* rr = rois + gp * 4;
    float ra = rr[0], rb_ = rr[1], rc = rr[2], rd = rr[3];
    float area_r = (rc - ra) * (rd - rb_);

    // IoU vs all gts: mask = any(iou > 0.5); idx = first argmax (invalid = -1)
    float best = -2.0f;
    int bidx = 0;
    float maskf = 0.f;
    for (int g = 0; g < MAX_GT; ++g) {
      float iou = -1.0f;
      if (g < cnt) {
        float g0 = sgt[g * 4 + 0], g1 = sgt[g * 4 + 1];
        float g2 = sgt[g * 4 + 2], g3 = sgt[g * 4 + 3];
        float area_g = (g2 - g0) * (g3 - g1);
        float ltx = fmaxf(g0, ra), lty = fmaxf(g1, rb_);
        float rbx = fminf(g2, rc), rby = fminf(g3, rd);
        float w = fmaxf(rbx - ltx, 0.f), h = fmaxf(rby - lty, 0.f);
        float inter = w * h;
        iou = inter / (area_g + area_r - inter);
      }
      if (iou > 0.5f) maskf = 1.0f;
      if (iou > best) { best = iou; bidx = g; }
    }

    // regression targets from matched gt
    float g0 = sgt[bidx * 4 + 0], g1 = sgt[bidx * 4 + 1];
    float g2 = sgt[bidx * 4 + 2], g3 = sgt[bidx * 4 + 3];
    float gx = (g2 + g0) * 0.5f, gy = (g3 + g1) * 0.5f;
    float gw = (g2 - g0) * 0.5f, gh = (g3 - g1) * 0.5f;
    float rx = (rc + ra) * 0.5f, ry = (rd + rb_) * 0.5f;
    float rw = (rc - ra) * 0.5f, rh = (rd - rb_) * 0.5f;
    float tx = (gx - rx) / (rw + 1e-8f);
    float ty = (gy - ry) / (rh + 1e-8f);
    float tw = logf(gw / (rw + 1e-8f));
    float th = logf(gh / (rh + 1e-8f));

    float l1 = sl1(r0 - tx) + sl1(r1 - ty) + sl1(r2 - tw) + sl1(r3 - th);

    acc_bg   += ce_bg;
    acc_fg   += ce_fg * maskf;
    acc_mask += maskf;
    acc_l1   += l1 * maskf * ps;
  }

  sred[tid * 4 + 0] = acc_bg;
  sred[tid * 4 + 1] = acc_fg;
  sred[tid * 4 + 2] = acc_mask;
  sred[tid * 4 + 3] = acc_l1;
  __syncthreads();
  for (int s = 128; s > 0; s >>= 1) {
    if (tid < s) {
#pragma unroll
      for (int j = 0; j < 4; ++j)
        sred[tid * 4 + j] += sred[(tid + s) * 4 + j];
    }
    __syncthreads();
  }
  if (tid == 0) {
#pragma unroll
    for (int j = 0; j < 4; ++j) part[b * 4 + j] = sred[j];
  }
}

// ---------------------------------------------------------------------------
// Kernel 3: combine per-batch partials into the scalar loss
// ---------------------------------------------------------------------------
__global__ void finalize(const float* __restrict__ part,
                         const int* __restrict__ gt_counts,
                         float* __restrict__ out) {
  if (threadIdx.x == 0 && blockIdx.x == 0) {
    float c_list = 0.f, c_bg = 0.f, bg = 0.f, l1_list = 0.f;
    for (int b = 0; b < BS; ++b) {
      float ce_bg_mean = part[b * 4 + 0] / (float)P;
      float c_fg = part[b * 4 + 1] / (part[b * 4 + 2] + 1e-7f);
      float l1 = part[b * 4 + 3] / (float)(BS * P);
      if (gt_counts[b] > 0) {
        c_list += c_fg;
        l1_list += l1;
      } else {
        c_bg += ce_bg_mean;
        bg += (float)P;
      }
    }
    out[0] = (c_list + c_bg / (bg + 1e-7f)) / ((float)BS + 1e-7f) + l1_list;
  }
}

// ---------------------------------------------------------------------------
extern "C" void kernel_launch(void* const* d_in, const int* in_sizes, int n_in,
                              void* d_out, int out_size, void* d_ws, size_t ws_size,
                              hipStream_t stream) {
  (void)in_sizes; (void)n_in; (void)out_size; (void)ws_size;
  const float* v         = (const float*)d_in[0];
  const float* gt_boxes  = (const float*)d_in[1];
  const int*   gt_counts = (const int*)  d_in[2];
  const float* rois      = (const float*)d_in[3];
  const float* labels    = (const float*)d_in[4];
  const float* pre_score = (const float*)d_in[5];
  const float* cls_w     = (const float*)d_in[6];
  const float* cls_b     = (const float*)d_in[7];
  const float* reg_w     = (const float*)d_in[8];
  const float* reg_b     = (const float*)d_in[9];

  float* cr   = (float*)d_ws;                 // [BS*P][8] logits = 2 MB
  float* part = cr + (size_t)BS * P * 8;      // [BS][4] partials

  gemm16<<<(BS * P) / (16 * 8), 256, 0, stream>>>(v, cls_w, cls_b, reg_w, reg_b, cr);
  loss_per_batch<<<BS, 256, 0, stream>>>(cr, gt_boxes, gt_counts, rois, labels,
                                         pre_score, part);
  finalize<<<1, 32, 0, stream>>>(part, gt_counts, (float*)d_out);
}